// FractalEmbedding_30709016167039
// MI455X (gfx1250) — compile-verified
//
#include <hip/hip_runtime.h>

typedef float v2f __attribute__((ext_vector_type(2)));
typedef float v8f __attribute__((ext_vector_type(8)));

#define STEPS 8
#define NF    16        // 2*STEPS features
#define EMBED 2048
#define NTOK  32768     // B*L = 4*8192

// ---------------------------------------------------------------------------
// Kernel 1: per-token Julia iteration -> 16 f32 features into workspace.
// One thread per token. Tiny traffic (2 MB write), fully parallel.
// ---------------------------------------------------------------------------
__global__ __launch_bounds__(256) void fractal_feats_kernel(
    const int*   __restrict__ token_ids,
    const float* __restrict__ c_real,
    const float* __restrict__ c_imag,
    float*       __restrict__ feats)
{
    int i = blockIdx.x * blockDim.x + threadIdx.x;
    if (i >= NTOK) return;
    int t = token_ids[i];
    float cr = c_real[t];
    float ci = c_imag[t];
    float zr = 0.0f, zi = 0.0f;
    float f[NF];
#pragma unroll
    for (int s = 0; s < STEPS; ++s) {
        float nzr = zr * zr - zi * zi + cr;
        float nzi = 2.0f * zr * zi + ci;
        zr = nzr; zi = nzi;
        f[2 * s]     = zr;
        f[2 * s + 1] = zi;
    }
    float4* dst = (float4*)(feats + (size_t)i * NF);
#pragma unroll
    for (int q = 0; q < 4; ++q)
        dst[q] = make_float4(f[4 * q], f[4 * q + 1], f[4 * q + 2], f[4 * q + 3]);
}

// ---------------------------------------------------------------------------
// Kernel 2: [N,16] x [16,2048] GEMM via V_WMMA_F32_16X16X4_F32.
// Block = 256 threads (8 waves). blockIdx.x = token tile (16 tokens).
// Each wave handles dim-tiles {wave, wave+8, ..., wave+120} (16 tiles).
// A (feats) loaded once per wave; 4 chained WMMAs (K=16) per dim tile.
// Output: one non-temporal f32 store per element (store-bandwidth bound).
// ---------------------------------------------------------------------------
__global__ __launch_bounds__(256) void fractal_proj_wmma_kernel(
    const float* __restrict__ feats,
    const float* __restrict__ proj_w,   // [EMBED, 16] row-major
    const float* __restrict__ scale_p,
    float*       __restrict__ out)      // [NTOK, EMBED]
{
    const int lane  = threadIdx.x & 31;
    const int wave  = threadIdx.x >> 5;
    const int ttile = blockIdx.x;          // 16-token tile index
    const int mrow  = lane & 15;           // M (for A) / N (for B) index
    const int khalf = (lane >> 4) * 2;     // K sub-offset: lanes16-31 hold K+2,K+3
    const float scale = *scale_p;

    // --- Load A tile: feats[ttile*16 + M][K], WMMA f32 16x16x4 A layout ---
    // op j uses K = 4j..4j+3; this lane supplies (K = 4j+khalf, 4j+khalf+1)
    const float* arow = feats + (size_t)(ttile * 16 + mrow) * NF + khalf;
    v2f a[4];
#pragma unroll
    for (int j = 0; j < 4; ++j)
        a[j] = *(const v2f*)(arow + 4 * j);

    const int moff = (lane >> 4) * 8;      // C/D: vgpr r -> M = r + 8*(lane>=16)

    for (int dt = wave; dt < EMBED / 16; dt += 8) {
        // --- Load B tile: B[K][N] = proj_w[dt*16 + N][K] ---
        const float* brow = proj_w + (size_t)(dt * 16 + mrow) * NF + khalf;
        v8f c = {};
#pragma unroll
        for (int j = 0; j < 4; ++j) {
            v2f b = *(const v2f*)(brow + 4 * j);
            // (neg_a, A, neg_b, B, c_mod, C, reuse_a, reuse_b)
            c = __builtin_amdgcn_wmma_f32_16x16x4_f32(
                    false, a[j], false, b, (short)0, c, false, false);
        }
        // --- Scale + non-temporal store of the 16x16 tile ---
        float* obase = out + (size_t)(ttile * 16) * EMBED + dt * 16 + mrow;
#pragma unroll
        for (int r = 0; r < 8; ++r) {
            __builtin_nontemporal_store(c[r] * scale,
                                        obase + (size_t)(moff + r) * EMBED);
        }
    }
}

// ---------------------------------------------------------------------------
extern "C" void kernel_launch(void* const* d_in, const int* in_sizes, int n_in,
                              void* d_out, int out_size, void* d_ws, size_t ws_size,
                              hipStream_t stream)
{
    const int*   token_ids = (const int*)  d_in[0];
    const float* c_real    = (const float*)d_in[1];
    const float* c_imag    = (const float*)d_in[2];
    const float* proj_w    = (const float*)d_in[3];
    const float* scale_p   = (const float*)d_in[4];
    float*       out       = (float*)d_out;
    float*       feats     = (float*)d_ws;   // NTOK*16 f32 = 2 MB scratch

    fractal_feats_kernel<<<NTOK / 256, 256, 0, stream>>>(
        token_ids, c_real, c_imag, feats);

    fractal_proj_wmma_kernel<<<NTOK / 16, 256, 0, stream>>>(
        feats, proj_w, scale_p, out);
}